// FineMatching_36945308680204
// MI455X (gfx1250) — compile-verified
//
#include <hip/hip_runtime.h>
#include <math.h>
#include <stdint.h>

// ---------------- problem constants (from reference) ----------------
#define WWIN   5
#define WW     25          // 5x5 window
#define CCH    128         // channels
#define ROW_F  132         // padded LDS row stride in floats (528B -> conflict-free b64 frags)
#define TROWS  32          // padded row count (25 real rows + 7 zeroed)
#define WAVES_PER_BLOCK 2  // 2 * (32*132 + 128) * 4B = 34816 B static LDS

#ifndef USE_ASYNC_COPY
#define USE_ASYNC_COPY 1   // CDNA5 global_load_async_to_lds_* path
#endif

typedef float v2f __attribute__((ext_vector_type(2)));
typedef float v4f __attribute__((ext_vector_type(4)));
typedef float v8f __attribute__((ext_vector_type(8)));

// SWAPX16 cross-half shuffle via ds_swizzle (group-of-32: xor=0x10, or=0, and=0x1f)
__device__ __forceinline__ float xor16(float x) {
  int i = __builtin_amdgcn_ds_swizzle(__float_as_int(x), 0x401F);
  return __int_as_float(i);
}

// grid helper: lin = {-1,-0.5,0,0.5,1}; gx = lin[r%5], gy = lin[r/5]
__device__ __forceinline__ float grid_x(int r) { return -1.0f + 0.5f * (float)(r % WWIN); }
__device__ __forceinline__ float grid_y(int r) { return -1.0f + 0.5f * (float)(r / WWIN); }

__global__ __launch_bounds__(WAVES_PER_BLOCK * 32)
void fine_matching_kernel(const float* __restrict__ f0,
                          const float* __restrict__ f1,
                          float* __restrict__ out, int M) {
  __shared__ __align__(16) float smem[WAVES_PER_BLOCK * (TROWS * ROW_F + CCH)];

  const int lane = threadIdx.x & 31;
  const int wave = threadIdx.x >> 5;
  const int m = blockIdx.x * WAVES_PER_BLOCK + wave;
  if (m >= M) return;                       // uniform per wave: EXEC stays all-ones

  float* tile = smem + wave * (TROWS * ROW_F + CCH);   // wave-private, padded tile
  float* f0s  = tile + TROWS * ROW_F;                  // picked feature vector

  const float* f1m = f1 + (size_t)m * (WW * CCH);
  const float* f0m = f0 + (size_t)m * (WW * CCH) + (WW / 2) * CCH;

  // ---------------- stage tile into LDS ----------------
#if USE_ASYNC_COPY
  // feat_picked: 128 floats = one b128 per lane (512B, fully coalesced)
  {
    uint32_t d = (uint32_t)(uintptr_t)(f0s + 4 * lane);
    uint64_t s = (uint64_t)(uintptr_t)(f0m + 4 * lane);
    asm volatile("global_load_async_to_lds_b128 %0, %1, off" :: "v"(d), "v"(s) : "memory");
  }
  // 25 rows of feat_f1: one coalesced 512B async b128 per row
  for (int r = 0; r < WW; ++r) {
    uint32_t d = (uint32_t)(uintptr_t)(tile + r * ROW_F + 4 * lane);
    uint64_t s = (uint64_t)(uintptr_t)(f1m + r * CCH + 4 * lane);
    asm volatile("global_load_async_to_lds_b128 %0, %1, off" :: "v"(d), "v"(s) : "memory");
  }
#else
  {
    v4f v = *(const v4f*)(f0m + 4 * lane);
    *(v4f*)(f0s + 4 * lane) = v;
  }
  for (int r = 0; r < WW; ++r) {
    v4f v = *(const v4f*)(f1m + r * CCH + 4 * lane);
    *(v4f*)(tile + r * ROW_F + 4 * lane) = v;
  }
#endif

  // zero pad rows 25..31 (keeps WMMA inputs finite; overlaps with async copies)
  {
    v4f z = {0.f, 0.f, 0.f, 0.f};
    for (int r = WW; r < TROWS; ++r)
      *(v4f*)(tile + r * ROW_F + 4 * lane) = z;
  }

#if USE_ASYNC_COPY
  asm volatile("s_wait_asynccnt 0" ::: "memory");
#endif

  // ---------------- sim via f32 WMMA, broadcast-B trick ----------------
  // A tile0 rows 0..15, A tile1 rows 16..31 (K = 128, accumulated over 32 k-blocks).
  // B[k][j] = f0s[4*kb+k] for ALL j  ->  C[i][j] = sim[i] replicated across columns.
  const int half = lane >> 4;                // 0: lanes 0-15, 1: lanes 16-31
  const int lrow = lane & 15;
  const float* a0p = tile + lrow * ROW_F + 2 * half;          // A rows 0..15
  const float* a1p = tile + (16 + lrow) * ROW_F + 2 * half;   // A rows 16..31
  const float* bp  = f0s + 2 * half;

  v8f acc0 = {0.f, 0.f, 0.f, 0.f, 0.f, 0.f, 0.f, 0.f};
  v8f acc1 = {0.f, 0.f, 0.f, 0.f, 0.f, 0.f, 0.f, 0.f};

  for (int kb = 0; kb < CCH / 4; ++kb) {
    v2f a0 = *(const v2f*)(a0p + 4 * kb);
    v2f a1 = *(const v2f*)(a1p + 4 * kb);
    v2f b  = *(const v2f*)(bp  + 4 * kb);
    acc0 = __builtin_amdgcn_wmma_f32_16x16x4_f32(false, a0, false, b,
                                                 (short)0, acc0, false, false);
    acc1 = __builtin_amdgcn_wmma_f32_16x16x4_f32(false, a1, false, b,
                                                 (short)0, acc1, false, false);
  }
  // Result (broadcast across columns):
  //   acc0[v] = sim[v + 8*half]        (rows 0..15, all valid)
  //   acc1[v] = sim[16 + v + 8*half]   (valid only while row < 25)

  // ---------------- softmax + spatial moments ----------------
  const float temp = 0.08838834764831845f;   // 1/sqrt(128)

  float mx = -INFINITY;
  #pragma unroll
  for (int v = 0; v < 8; ++v) mx = fmaxf(mx, acc0[v]);
  #pragma unroll
  for (int v = 0; v < 8; ++v) {
    int r = 16 + v + 8 * half;
    if (r < WW) mx = fmaxf(mx, acc1[v]);
  }
  mx = fmaxf(mx, xor16(mx));                 // combine lane halves

  float sum = 0.f, cx = 0.f, cy = 0.f, sxx = 0.f, syy = 0.f;
  #pragma unroll
  for (int v = 0; v < 8; ++v) {
    int r = v + 8 * half;
    float e = __expf(temp * (acc0[v] - mx));
    float gx = grid_x(r), gy = grid_y(r);
    sum += e; cx += e * gx; cy += e * gy; sxx += e * gx * gx; syy += e * gy * gy;
  }
  #pragma unroll
  for (int v = 0; v < 8; ++v) {
    int r = 16 + v + 8 * half;
    if (r < WW) {
      float e = __expf(temp * (acc1[v] - mx));
      float gx = grid_x(r), gy = grid_y(r);
      sum += e; cx += e * gx; cy += e * gy; sxx += e * gx * gx; syy += e * gy * gy;
    }
  }
  sum += xor16(sum);
  cx  += xor16(cx);
  cy  += xor16(cy);
  sxx += xor16(sxx);
  syy += xor16(syy);

  float inv = 1.0f / sum;
  cx *= inv; cy *= inv; sxx *= inv; syy *= inv;
  float vx = sxx - cx * cx;
  float vy = syy - cy * cy;
  float stdv = sqrtf(fmaxf(vx, 1e-10f)) + sqrtf(fmaxf(vy, 1e-10f));

  if (lane == 0) {
    float* o = out + 3 * (size_t)m;
    o[0] = cx; o[1] = cy; o[2] = stdv;
  }
}

extern "C" void kernel_launch(void* const* d_in, const int* in_sizes, int n_in,
                              void* d_out, int out_size, void* d_ws, size_t ws_size,
                              hipStream_t stream) {
  (void)n_in; (void)d_ws; (void)ws_size; (void)out_size;
  const float* f0 = (const float*)d_in[0];
  const float* f1 = (const float*)d_in[1];
  float* out = (float*)d_out;
  int M = in_sizes[0] / (WW * CCH);          // 50000
  dim3 block(WAVES_PER_BLOCK * 32);
  dim3 grid((M + WAVES_PER_BLOCK - 1) / WAVES_PER_BLOCK);
  fine_matching_kernel<<<grid, block, 0, stream>>>(f0, f1, out, M);
}